// TextureMartingaleModule_2199023256201
// MI455X (gfx1250) — compile-verified
//
#include <hip/hip_runtime.h>
#include <stdint.h>

#define IMG_H 512
#define IMG_W 512
#define HWPLANE (IMG_H * IMG_W)
#define EPSF 1e-6f
#define EXP_NEG_HALF 0.6065306597126334f

// Block geometry: 8 waves, each wave owns one 16x16 output tile (2 rows x 4 cols of tiles)
#define TILES_X 4
#define TILES_Y 2
#define BLK_W (TILES_X * 16)   // 64
#define BLK_H (TILES_Y * 16)   // 32
#define HALO 4                 // radius of largest window (9)
#define EXT_W (BLK_W + 2*HALO) // 72
#define EXT_H (BLK_H + 2*HALO) // 40
#define P_STRIDE 73            // odd stride to spread LDS banks
#define H_STRIDE 65

typedef float v2f __attribute__((ext_vector_type(2)));
typedef float v8f __attribute__((ext_vector_type(8)));

// One window size, fully compile-time: horizontal box sums (block-cooperative),
// vertical box sums as band-matrix WMMA (per-wave), then the non-separable
// sum|p-mean| pass (register-column reuse across the lane's 8 pixels) + features.
template<int Wn>
__device__ __forceinline__ void process_window(
    int widx,
    float (*lds_p)[P_STRIDE], float (*lds_plog)[P_STRIDE], float (*lds_h)[H_STRIDE],
    int tid, int ty0, int tx0, int hl, int ln,
    int by0, int bx0, int bc, float* __restrict__ out)
{
    constexpr int R  = Wn / 2;
    constexpr float Kf = (float)(Wn * Wn);

    const int hc = tid & 63;           // H-pass column (0..63)
    const int he = tid >> 6;           // H-pass row group (0..3)

    v8f acc0 = {}, acc1 = {}, acc2 = {};

#pragma unroll
    for (int q = 0; q < 3; ++q) {
        // ---- horizontal window sums into lds_h: fixed mapping, 10 iters/thread ----
#pragma unroll
        for (int i = 0; i < EXT_H / 4; ++i) {
            int e = he + 4 * i;
            int base = hc + HALO - R;
            float s = 0.0f;
            if (q == 0) {
#pragma unroll
                for (int dx = 0; dx < Wn; ++dx) s += lds_p[e][base + dx];
            } else if (q == 1) {
#pragma unroll
                for (int dx = 0; dx < Wn; ++dx) { float v = lds_p[e][base + dx]; s += v * v; }
            } else {
#pragma unroll
                for (int dx = 0; dx < Wn; ++dx) s += lds_plog[e][base + dx];
            }
            lds_h[e][hc] = s;
        }
        __syncthreads();

        // ---- vertical sums via WMMA: S(16x16) = Band(16x24) x H(24x16), K=24 ----
        v8f c8 = {0.f, 0.f, 0.f, 0.f, 0.f, 0.f, 0.f, 0.f};
#if __has_builtin(__builtin_amdgcn_wmma_f32_16x16x4_f32)
#pragma unroll
        for (int s = 0; s < 6; ++s) {
            // A 16x4 f32 layout: lane -> M = ln; VGPR v -> K = v + 2*hl
            int k0 = 4 * s + 2 * hl;
            int k1 = k0 + 1;
            v2f a, b;
            a.x = (__builtin_abs(k0 - (ln + HALO)) <= R) ? 1.0f : 0.0f;
            a.y = (__builtin_abs(k1 - (ln + HALO)) <= R) ? 1.0f : 0.0f;
            // B 4x16 f32 layout (mirrors A): lane -> N = ln; VGPR v -> K = v + 2*hl
            b.x = lds_h[ty0 + k0][tx0 + ln];
            b.y = lds_h[ty0 + k1][tx0 + ln];
            c8 = __builtin_amdgcn_wmma_f32_16x16x4_f32(
                    false, a, false, b, (short)0, c8, false, false);
        }
#else
#pragma unroll
        for (int rr = 0; rr < 8; ++rr) {
            int M = rr + 8 * hl;
            float s = 0.0f;
#pragma unroll
            for (int dy = -R; dy <= R; ++dy)
                s += lds_h[ty0 + M + HALO + dy][tx0 + ln];
            c8[rr] = s;
        }
#endif
        if (q == 0) acc0 = c8; else if (q == 1) acc1 = c8; else acc2 = c8;
        __syncthreads();   // lds_h reused by next q / next window
    }

    // ---- per-pixel pass: sum |p - mean| with register-column reuse ----
    // The lane's 8 pixels are (M = rr + 8*hl, N = ln), rr = 0..7 -- one column,
    // consecutive rows. For a fixed dx their w x w windows touch only (Wn+7)
    // distinct rows, so load that column once and reuse it for all 8 pixels.
    float m[8], sabs[8];
#pragma unroll
    for (int rr = 0; rr < 8; ++rr) { m[rr] = acc0[rr] / Kf; sabs[rr] = 0.0f; }

    const int pe_base = ty0 + 8 * hl + HALO - R;   // first window row touched
    const int pc0     = tx0 + ln + HALO - R;       // first window col touched

    for (int dx = 0; dx < Wn; ++dx) {
        float col[Wn + 7];
#pragma unroll
        for (int j = 0; j < Wn + 7; ++j) col[j] = lds_p[pe_base + j][pc0 + dx];
#pragma unroll
        for (int rr = 0; rr < 8; ++rr) {
#pragma unroll
            for (int dy = 0; dy < Wn; ++dy)
                sabs[rr] += fabsf(col[rr + dy] - m[rr]);
        }
    }

    // ---- feature math + stores ----
#pragma unroll
    for (int rr = 0; rr < 8; ++rr) {
        int M  = rr + 8 * hl;
        float S1 = acc0[rr], S2 = acc1[rr], SL = acc2[rr];
        float mm = m[rr];

        // torch.std is unbiased (ddof=1); eps added to std, then normed^2 averaged
        float bv   = S2 / Kf - mm * mm;                        // biased variance
        float stdv = sqrtf(fmaxf(bv * (Kf / (Kf - 1.0f)), 0.0f)) + EPSF;
        float contrast = bv / (stdv * stdv);
        float energy   = S2 / Kf;
        float entropy  = -SL / Kf;
        float homog    = 1.0f / (1.0f + sabs[rr] / Kf);

        int gy = by0 + ty0 + M;
        int gx = bx0 + tx0 + ln;
        size_t pix = (size_t)gy * IMG_W + gx;
        size_t pb  = ((size_t)bc * 16 + widx * 4) * HWPLANE + pix;
        // M = exp(theta*log(f+eps) - 0.5*theta^2), theta=1  ->  (f+eps)*e^{-1/2}
        out[pb]                     = (contrast + EPSF) * EXP_NEG_HALF;
        out[pb + (size_t)HWPLANE]   = (energy   + EPSF) * EXP_NEG_HALF;
        out[pb + 2*(size_t)HWPLANE] = (entropy  + EPSF) * EXP_NEG_HALF;
        out[pb + 3*(size_t)HWPLANE] = (homog    + EPSF) * EXP_NEG_HALF;
    }
}

__global__ __launch_bounds__(256) void glcm_texture_kernel(const float* __restrict__ x,
                                                           float* __restrict__ out)
{
    __shared__ float lds_p[EXT_H][P_STRIDE];     // staged input tile (+halo, zero padded)
    __shared__ float lds_plog[EXT_H][P_STRIDE];  // p * log(p + eps)
    __shared__ float lds_h[EXT_H][H_STRIDE];     // horizontal window sums (reused per q,w)

    const int tid = threadIdx.x;
    const int bc  = blockIdx.z;                  // b*C + c
    const int by0 = blockIdx.y * BLK_H;
    const int bx0 = blockIdx.x * BLK_W;
    const float* __restrict__ src = x + (size_t)bc * HWPLANE;

    // ---------------- Stage haloed tile into LDS ----------------
    const bool interior = (by0 >= HALO) && (by0 + BLK_H + HALO <= IMG_H) &&
                          (bx0 >= HALO) && (bx0 + BLK_W + HALO <= IMG_W);
    if (interior) {
        // CDNA5 async global->LDS copy: bypasses VGPRs, tracked by ASYNCcnt.
        uint64_t sbase = (uint64_t)src;
        for (int idx = tid; idx < EXT_H * EXT_W; idx += 256) {
            int e = idx / EXT_W, c = idx % EXT_W;
            uint32_t ldsoff = (uint32_t)(uintptr_t)&lds_p[e][c]; // low 32 bits = LDS offset
            uint32_t goff   = (uint32_t)(((by0 - HALO + e) * IMG_W + (bx0 - HALO + c)) * 4);
            asm volatile("global_load_async_to_lds_b32 %0, %1, %2"
                         :: "v"(ldsoff), "v"(goff), "s"(sbase) : "memory");
        }
        asm volatile("s_wait_asynccnt 0x0" ::: "memory");
    } else {
        for (int idx = tid; idx < EXT_H * EXT_W; idx += 256) {
            int e = idx / EXT_W, c = idx % EXT_W;
            int gy = by0 - HALO + e, gx = bx0 - HALO + c;
            float v = 0.0f;
            if (gy >= 0 && gy < IMG_H && gx >= 0 && gx < IMG_W) v = src[gy * IMG_W + gx];
            lds_p[e][c] = v;
        }
    }
    __syncthreads();

    // p * log(p + eps) field (log computed once per element, reused by all 4 windows)
    for (int idx = tid; idx < EXT_H * EXT_W; idx += 256) {
        int e = idx / EXT_W, c = idx % EXT_W;
        float v = lds_p[e][c];
        lds_plog[e][c] = v * __logf(v + EPSF);
    }
    __syncthreads();

    // ---------------- Per-wave tile identification ----------------
    const int lane = tid & 31;
    const int wid  = tid >> 5;
    const int ty0  = (wid >> 2) * 16;  // tile origin inside block (output coords)
    const int tx0  = (wid & 3) * 16;
    const int hl   = lane >> 4;        // lane half (0/1)
    const int ln   = lane & 15;

    process_window<3>(0, lds_p, lds_plog, lds_h, tid, ty0, tx0, hl, ln, by0, bx0, bc, out);
    process_window<5>(1, lds_p, lds_plog, lds_h, tid, ty0, tx0, hl, ln, by0, bx0, bc, out);
    process_window<7>(2, lds_p, lds_plog, lds_h, tid, ty0, tx0, hl, ln, by0, bx0, bc, out);
    process_window<9>(3, lds_p, lds_plog, lds_h, tid, ty0, tx0, hl, ln, by0, bx0, bc, out);
}

extern "C" void kernel_launch(void* const* d_in, const int* in_sizes, int n_in,
                              void* d_out, int out_size, void* d_ws, size_t ws_size,
                              hipStream_t stream) {
    (void)n_in; (void)out_size; (void)d_ws; (void)ws_size;
    const float* x = (const float*)d_in[0];
    float* out = (float*)d_out;
    int planes = in_sizes[0] / HWPLANE;          // B*C = 6
    dim3 grid(IMG_W / BLK_W, IMG_H / BLK_H, planes);
    glcm_texture_kernel<<<grid, 256, 0, stream>>>(x, out);
}